// KMeansCluster_61684320305432
// MI455X (gfx1250) — compile-verified
//
#include <hip/hip_runtime.h>

#define K_CL     64
#define D_DIM    64
#define N_PTS    65536
#define B_BATCH  8
#define MAX_IT   20
#define TOL_F    0.005f
#define LDS_PAD  68   // centers row stride in floats (bank-conflict padding)

typedef __attribute__((ext_vector_type(2))) float v2f;
typedef __attribute__((ext_vector_type(8))) float v8f;

// ---------------------------------------------------------------------------
// Init: gather initial centers, zero accumulators, clear convergence flags.
// ---------------------------------------------------------------------------
__global__ void km_init(const float* __restrict__ feats,
                        const int* __restrict__ init_idx,
                        float* __restrict__ centers,
                        float* __restrict__ sums,
                        float* __restrict__ counts,
                        int* __restrict__ flags) {
  const int b = blockIdx.x;
  const float* fb = feats + (size_t)b * N_PTS * D_DIM;
  float* cb = centers + b * K_CL * D_DIM;
  float* sb = sums    + b * K_CL * D_DIM;
  for (int i = threadIdx.x; i < K_CL * D_DIM; i += blockDim.x) {
    const int k = i >> 6, d = i & 63;
    const int src = init_idx[b * K_CL + k];
    cb[i] = fb[(size_t)src * D_DIM + d];
    sb[i] = 0.0f;
  }
  for (int i = threadIdx.x; i < K_CL; i += blockDim.x) counts[b * K_CL + i] = 0.0f;
  if (threadIdx.x == 0) flags[b] = 0;
}

// ---------------------------------------------------------------------------
// Assign + accumulate. Block = 256 threads = 8 waves; each wave owns 16
// tiles of 16 points -> 2048 points/block, 32 blocks/batch.
// Distance matmul via V_WMMA_F32_16X16X4_F32:
//   A = centers tile (16 clusters x 4 dims per step, from LDS)
//   B = features tile (4 dims x 16 points, from global)
//   D accum = -? we accumulate dot, epilogue applies  c2[k] - 2*dot.
// ---------------------------------------------------------------------------
__launch_bounds__(256, 1)
__global__ void km_assign(const float* __restrict__ feats,
                          const float* __restrict__ centers,
                          float* __restrict__ sums,
                          float* __restrict__ counts,
                          const int* __restrict__ flags,
                          int* __restrict__ out) {
  const int b = blockIdx.y;
  if (flags[b]) return;   // converged: keep previous assignments/centers

  __shared__ float sC[K_CL * LDS_PAD];   // centers, padded rows
  __shared__ float sC2[K_CL];            // |c_k|^2
  __shared__ float sSum[K_CL * D_DIM];   // per-block partial sums
  __shared__ float sCnt[K_CL];           // per-block partial counts

  const float* fb = feats   + (size_t)b * N_PTS * D_DIM;
  const float* cb = centers + b * K_CL * D_DIM;
  const int tid = threadIdx.x;

  for (int i = tid; i < K_CL * D_DIM; i += 256) {
    sC[(i >> 6) * LDS_PAD + (i & 63)] = cb[i];
    sSum[i] = 0.0f;
  }
  for (int i = tid; i < K_CL; i += 256) sCnt[i] = 0.0f;
  __syncthreads();

  if (tid < K_CL) {
    float acc = 0.0f;
    for (int d = 0; d < D_DIM; ++d) {
      const float v = sC[tid * LDS_PAD + d];
      acc += v * v;
    }
    sC2[tid] = acc;
  }
  __syncthreads();

  const int wave = tid >> 5;
  const int lane = tid & 31;
  const int lrow = lane & 15;   // point (B/N column) / cluster row within half
  const int lhi  = lane >> 4;   // half-wave select (K offset 0 or 2; M offset 0 or 8)

  for (int t = 0; t < 16; ++t) {
    const int p0 = blockIdx.x * 2048 + wave * 256 + t * 16;

    // Prefetch next tile's features (gfx1250 global_prefetch path).
    if (t + 1 < 16)
      __builtin_prefetch(fb + (size_t)(p0 + 16 + lrow) * D_DIM, 0, 1);

    // --- load all 16 B fragments (feature tile, transposed into WMMA layout)
    // lane layout for 32-bit B 4x16: lanes 0-15 hold (K=0,K=1) for point=lane,
    // lanes 16-31 hold (K=2,K=3) for point=lane-16.
    v2f bf[16];
    const float* fp = fb + (size_t)(p0 + lrow) * D_DIM + lhi * 2;
#pragma unroll
    for (int s = 0; s < 16; ++s)
      bf[s] = *(const v2f*)(fp + s * 4);

    // --- 4 cluster tiles x 16 k-steps of fp32 WMMA
    v8f acc[4];
#pragma unroll
    for (int ct = 0; ct < 4; ++ct) {
      v8f c = {};
      const float* arow = &sC[(ct * 16 + lrow) * LDS_PAD + lhi * 2];
#pragma unroll
      for (int s = 0; s < 16; ++s) {
        v2f a = { arow[s * 4], arow[s * 4 + 1] };
        c = __builtin_amdgcn_wmma_f32_16x16x4_f32(
                /*neg_a=*/false, a, /*neg_b=*/false, bf[s],
                /*c_mod=*/(short)0, c, /*reuse_a=*/false, /*reuse_b=*/false);
      }
      acc[ct] = c;
    }

    // --- epilogue: score = c2[k] - 2*dot ; local argmin then lane^16 combine
    float best = 3.4e38f;
    int bestk = 0;
#pragma unroll
    for (int ct = 0; ct < 4; ++ct) {
      const int base = ct * 16 + lhi * 8;   // C/D layout: upper half-wave = M+8
#pragma unroll
      for (int r = 0; r < 8; ++r) {
        const float sc = sC2[base + r] - 2.0f * acc[ct][r];
        if (sc < best) { best = sc; bestk = base + r; }
      }
    }
    {
      const float ob = __shfl_xor(best, 16, 32);
      const int   ok = __shfl_xor(bestk, 16, 32);
      if (ob < best || (ob == best && ok < bestk)) { best = ob; bestk = ok; }
    }

    if (lane < 16) {
      out[b * N_PTS + p0 + lrow] = bestk;
      atomicAdd(&sCnt[bestk], 1.0f);
    }

    // --- accumulate feature sums: broadcast cluster of point p, lanes cover
    // 2 dims each (banks 0..63, conflict-free ds_add_f32).
#pragma unroll 4
    for (int p = 0; p < 16; ++p) {
      const int kp = __shfl(bestk, p, 32);
      const v2f f = *(const v2f*)(fb + (size_t)(p0 + p) * D_DIM + lane * 2);
      atomicAdd(&sSum[kp * D_DIM + lane * 2 + 0], f.x);
      atomicAdd(&sSum[kp * D_DIM + lane * 2 + 1], f.y);
    }
  }

  __syncthreads();
  float* gs = sums + b * K_CL * D_DIM;
  for (int i = tid; i < K_CL * D_DIM; i += 256) atomicAdd(&gs[i], sSum[i]);
  for (int i = tid; i < K_CL; i += 256) atomicAdd(&counts[b * K_CL + i], sCnt[i]);
}

// ---------------------------------------------------------------------------
// Update: centers <- sums/counts (keep old if empty), shift = sum_k ||dc_k||,
// set converged flag, zero accumulators for next iteration.
// One block per batch, 64 threads (one per cluster).
// ---------------------------------------------------------------------------
__global__ void km_update(float* __restrict__ centers,
                          float* __restrict__ sums,
                          float* __restrict__ counts,
                          int* __restrict__ flags) {
  const int b = blockIdx.x;
  if (flags[b]) return;
  __shared__ float red[K_CL];
  const int k = threadIdx.x;

  const float cnt = counts[b * K_CL + k];
  float* cr = centers + (size_t)(b * K_CL + k) * D_DIM;
  float* sr = sums    + (size_t)(b * K_CL + k) * D_DIM;

  float shift2 = 0.0f;
  for (int d = 0; d < D_DIM; ++d) {
    const float oldc = cr[d];
    const float newc = (cnt > 0.5f) ? (sr[d] / fmaxf(cnt, 1.0f)) : oldc;
    const float df = newc - oldc;
    shift2 += df * df;
    cr[d] = newc;
    sr[d] = 0.0f;
  }
  counts[b * K_CL + k] = 0.0f;

  red[k] = sqrtf(shift2);
  __syncthreads();
  for (int off = 32; off > 0; off >>= 1) {
    if (k < off) red[k] += red[k + off];
    __syncthreads();
  }
  if (k == 0 && red[0] < TOL_F * (float)N_PTS) flags[b] = 1;
}

// ---------------------------------------------------------------------------
extern "C" void kernel_launch(void* const* d_in, const int* in_sizes, int n_in,
                              void* d_out, int out_size, void* d_ws, size_t ws_size,
                              hipStream_t stream) {
  (void)in_sizes; (void)n_in; (void)out_size; (void)ws_size;
  const float* feats    = (const float*)d_in[0];
  const int*   init_idx = (const int*)d_in[1];
  int*         out      = (int*)d_out;

  float* centers = (float*)d_ws;                       // B*K*D
  float* sums    = centers + B_BATCH * K_CL * D_DIM;   // B*K*D
  float* counts  = sums    + B_BATCH * K_CL * D_DIM;   // B*K
  int*   flags   = (int*)(counts + B_BATCH * K_CL);    // B

  km_init<<<B_BATCH, 256, 0, stream>>>(feats, init_idx, centers, sums, counts, flags);
  for (int it = 0; it < MAX_IT; ++it) {
    km_assign<<<dim3(32, B_BATCH), 256, 0, stream>>>(feats, centers, sums, counts,
                                                     flags, out);
    km_update<<<B_BATCH, K_CL, 0, stream>>>(centers, sums, counts, flags);
  }
}